// CMELossAngularProfileMSE_V2_84937273245912
// MI455X (gfx1250) — compile-verified
//
#include <hip/hip_runtime.h>
#include <stdint.h>

typedef __attribute__((ext_vector_type(2))) float v2f;
typedef __attribute__((ext_vector_type(8))) float v8f;

#define THETA      360
#define THETA_PAD  368                  // 23 * 16
#define NCOLT      23                   // column tiles of 16 thetas
#define RDIM       2048
#define BATCH      128
#define KDIM       2
#define CHUNKS     16
#define CHUNK_ROWS (RDIM / CHUNKS)      // 128
#define TILE_ROWS  16
#define TILES      (CHUNK_ROWS / TILE_ROWS) // 8
#define NTHREADS   96                   // 3 waves; 1440 xfers/tile / 96 = 15 exact
#define XFERS_PER_THREAD 15             // (16 rows * 90 float4) / 96
#define BT         (BATCH * THETA)      // 46080
#define NBLK_EPI   90                   // 46080 / 512 exactly
#define SIGMA      10.0f
#define ALPHA_W    2.0f

// ---------------------------------------------------------------------------
// Async copy of one contiguous 16-row x 360-float tile (23040 B) into a
// padded LDS tile (row stride 368 floats) using gfx1250 async DMA.
// Exactly 15 b128 transfers per thread -> 15 ASYNCcnt increments per wave.
// ---------------------------------------------------------------------------
__device__ __forceinline__ void copy_tile_async(const float* __restrict__ gbase,
                                                float (*buf)[THETA_PAD],
                                                int tid) {
#pragma unroll
    for (int it = 0; it < XFERS_PER_THREAD; ++it) {
        int f   = it * NTHREADS + tid;     // 0..1439 flat float4 index
        int row = f / 90;
        int col = f - row * 90;            // float4 column within the row
        const float* g = gbase + row * THETA + col * 4;
        uint32_t lds_off = (uint32_t)(uintptr_t)(&buf[row][col * 4]);
        uint64_t gaddr   = (uint64_t)(uintptr_t)g;
        asm volatile("global_load_async_to_lds_b128 %0, %1, off"
                     :: "v"(lds_off), "v"(gaddr)
                     : "memory");
    }
}

// ---------------------------------------------------------------------------
// Kernel A: per-chunk radial sums  P[chunk][b*360+theta] = sum_{r in chunk} mask
// Double-buffered async global->LDS, WMMA ones-trick column-sum reduction.
// Each output slot is written exactly once -> no init, no atomics.
// ---------------------------------------------------------------------------
__global__ void __launch_bounds__(NTHREADS)
cme_radial_sum_kernel(const float* __restrict__ mask, float* __restrict__ P) {
    __shared__ float buf[2][TILE_ROWS][THETA_PAD];

    const int tid   = threadIdx.x;
    const int lane  = tid & 31;
    const int wave  = tid >> 5;            // 0..2
    const int blk   = blockIdx.x;
    const int b     = blk / CHUNKS;
    const int chunk = blk - b * CHUNKS;
    const int r0    = chunk * CHUNK_ROWS;
    const float* base = mask + ((size_t)b * RDIM + (size_t)r0) * THETA;

    // Zero the 8-float pad columns (360..367) of every row, both buffers.
    if (tid < 32) {
        int pb = tid >> 4;                 // buffer 0/1
        int pr = tid & 15;                 // row
#pragma unroll
        for (int i = 0; i < 8; ++i) buf[pb][pr][THETA + i] = 0.0f;
    }

    // Per-wave WMMA accumulators: wave w owns column tiles {w, w+3, w+6, ...}
    v8f acc[8];
#pragma unroll
    for (int j = 0; j < 8; ++j)
        acc[j] = (v8f){0.f, 0.f, 0.f, 0.f, 0.f, 0.f, 0.f, 0.f};

    v2f ones = {1.0f, 1.0f};               // all-ones 16x4 A-matrix

    const int nloc = lane & 15;            // N (theta within column tile)
    const int rsel = ((lane >> 4) & 1) << 1; // 0 or 2: K rows held by this half

    copy_tile_async(base, buf[0], tid);

    for (int t = 0; t < TILES; ++t) {
        if (t + 1 < TILES) {
            copy_tile_async(base + (size_t)(t + 1) * TILE_ROWS * THETA,
                            buf[(t + 1) & 1], tid);
            // 15 newer ops may remain outstanding; the 15 for tile t are done.
            asm volatile("s_wait_asynccnt 15" ::: "memory");
        } else {
            asm volatile("s_wait_asynccnt 0" ::: "memory");
        }
        __syncthreads();   // all waves' DMA slices visible in LDS

        const float (*cb)[THETA_PAD] = buf[t & 1];
#pragma unroll
        for (int j = 0; j < 8; ++j) {
            int ct = wave + 3 * j;         // wave-uniform: EXEC all-ones or execz-skip
            if (ct < NCOLT) {
#pragma unroll
                for (int kc = 0; kc < 4; ++kc) {
                    int rr = kc * 4 + rsel;
                    v2f bb;
                    bb[0] = cb[rr][ct * 16 + nloc];
                    bb[1] = cb[rr + 1][ct * 16 + nloc];
                    // D[m,n] += sum_k B[k,n]  (A == ones => K-slot permutation
                    // in the B layout cannot change the result)
                    acc[j] = __builtin_amdgcn_wmma_f32_16x16x4_f32(
                        false, ones, false, bb, (short)0, acc[j], false, false);
                }
            }
        }
        __syncthreads();   // done reading buf[t&1] before tile t+2 overwrites it
    }

    // Extract row M=0 of each accumulator: element 0, lanes 0..15 = N 0..15.
#pragma unroll
    for (int j = 0; j < 8; ++j) {
        int ct = wave + 3 * j;
        if (ct < NCOLT) {
            int theta = ct * 16 + nloc;
            if (lane < 16 && theta < THETA) {
                P[(size_t)chunk * BT + b * THETA + theta] = acc[j][0];
            }
        }
    }
}

// ---------------------------------------------------------------------------
// Kernel B: combine chunk partials (fixed order) + Gaussian target, weight,
// weighted MSE; fixed-order block tree reduction -> one partial per block.
// ---------------------------------------------------------------------------
__global__ void __launch_bounds__(512)
cme_epilogue_kernel(const float* __restrict__ P,
                    const int* __restrict__ tmin_i,
                    const int* __restrict__ tmax_i,
                    float* __restrict__ S) {
    __shared__ float red[512];
    const int tid = threadIdx.x;
    const int idx = blockIdx.x * 512 + tid;      // 0..46079, exact cover

    // Deterministic ordered sum of the 16 chunk partials
    float s = 0.0f;
#pragma unroll
    for (int c = 0; c < CHUNKS; ++c) s += P[(size_t)c * BT + idx];
    float a = s * (1.0f / (float)RDIM);          // radial mean

    const int b  = idx / THETA;
    const int th = idx - b * THETA;
    const float thf = (float)th;

    float T = 0.0f;
    float dmax = 0.0f;
#pragma unroll
    for (int k = 0; k < KDIM; ++k) {
        float tn = (float)tmin_i[b * KDIM + k];
        float tx = (float)tmax_i[b * KDIM + k];

        // Gaussian center with wrap handling
        float ct;
        if (tn <= tx) {
            ct = 0.5f * (tn + tx);
        } else {
            ct = 0.5f * (tn + tx + 360.0f);
            if (ct >= 360.0f) ct -= 360.0f;
        }
        float d = fabsf(thf - ct);
        d = fminf(d, 360.0f - d);
        T += expf(-0.5f * (d * d) * (1.0f / (SIGMA * SIGMA)));

        // weight center: (tmin + (tmax - tmin) mod 360) / 2
        float m = tx - tn;
        if (m < 0.0f) m += 360.0f;
        float cw = 0.5f * (tn + m);
        float dw = fabsf(thf - cw);
        dw = fminf(dw, 360.0f - dw);
        dmax = fmaxf(dmax, dw);
    }
    T = fminf(T, 1.0f);
    float w = 1.0f + ALPHA_W * dmax * (1.0f / 180.0f);
    float diff = a - T;

    red[tid] = diff * diff * w;
    __syncthreads();
    // Fixed-order tree reduction (deterministic)
#pragma unroll
    for (int off = 256; off > 0; off >>= 1) {
        if (tid < off) red[tid] += red[tid + off];
        __syncthreads();
    }
    if (tid == 0) S[blockIdx.x] = red[0];
}

// ---------------------------------------------------------------------------
// Kernel C: deterministic final scalar: sequential sum of 90 block partials.
// ---------------------------------------------------------------------------
__global__ void __launch_bounds__(128)
cme_final_reduce_kernel(const float* __restrict__ S, float* __restrict__ out) {
    __shared__ float v[NBLK_EPI];
    const int tid = threadIdx.x;
    if (tid < NBLK_EPI) v[tid] = S[tid];
    __syncthreads();
    if (tid == 0) {
        float t = 0.0f;
        for (int i = 0; i < NBLK_EPI; ++i) t += v[i];
        out[0] = t * (1.0f / (float)BT);   // mean over theta, then batch; LAMBDA=1
    }
}

// ---------------------------------------------------------------------------
extern "C" void kernel_launch(void* const* d_in, const int* in_sizes, int n_in,
                              void* d_out, int out_size, void* d_ws, size_t ws_size,
                              hipStream_t stream) {
    const float* mask = (const float*)d_in[0];   // [128,1,2048,360] fp32
    const int*   tmin = (const int*)d_in[1];     // [128,2] int32
    const int*   tmax = (const int*)d_in[2];     // [128,2] int32
    float* out = (float*)d_out;                  // scalar

    float* P = (float*)d_ws;                     // [16][46080] chunk partials
    float* S = P + (size_t)CHUNKS * BT;          // [90] block partials

    cme_radial_sum_kernel<<<BATCH * CHUNKS, NTHREADS, 0, stream>>>(mask, P);
    cme_epilogue_kernel<<<NBLK_EPI, 512, 0, stream>>>(P, tmin, tmax, S);
    cme_final_reduce_kernel<<<1, 128, 0, stream>>>(S, out);
}